// BLS_33466385170439
// MI455X (gfx1250) — compile-verified
//
#include <hip/hip_runtime.h>

// ---------------------------------------------------------------------------
// Problem constants (from reference)
// ---------------------------------------------------------------------------
#define N1    192
#define N2    48
#define D_IN  1024
#define DP1   1025
#define D_OUT 512
#define LAM   0.001f
#define ITRS  100
#define LR    0.01f

#define GRP_ELEMS (DP1 * N1)          // 196800 elements per group matrix
#define TOT_W     (N2 * GRP_ELEMS)    // 9,446,400

typedef float v2f __attribute__((ext_vector_type(2)));
typedef float v8f __attribute__((ext_vector_type(8)));

// ---------------------------------------------------------------------------
// Stage 1: h1^T * M matvecs for We (-> A_raw partials) and W0 (-> p partials),
// plus ||W0||_F^2 partials, using V_WMMA_F32_16X16X4_F32.
//
// One wave (32 threads) per block. blockIdx.x = n*24 + mat*12 + tile,
// blockIdx.y = d-split j in [0,4). Each wave walks 256 d-values (64 WMMAs),
// accumulating 16 output columns in D row 0 (A rows 1..15 are zero, so D
// rows 1..15 contribute nothing).
//
// A-tile (16x4 f32) per ISA layout: lanes 0-15 hold K=0,1 in v[0],v[1];
// lanes 16-31 hold K=2,3; M = lane%16. Row 0 only => lane0 and lane16 carry
// h1 values, all other lanes zero (built with v_cndmask, no EXEC change).
// B-tile (4x16 f32): lanes 0-15 v[0],v[1] = rows K=0,1 (N=lane);
// lanes 16-31 = rows K=2,3 (N=lane-16).
// ---------------------------------------------------------------------------
__global__ __launch_bounds__(32) void bls_matvec_wmma(
    const float* __restrict__ x,
    const float* __restrict__ We,
    const float* __restrict__ W0,
    float* __restrict__ partA,   // [4][48][192]
    float* __restrict__ partP,   // [4][48][192]
    float* __restrict__ s0p)     // [48][48]  (tile*4 + j per group)
{
    const int lane = threadIdx.x;            // 0..31
    const int bx   = blockIdx.x;             // 0..1151
    const int j    = blockIdx.y;             // 0..3
    const int n    = bx / 24;
    const int rem  = bx % 24;
    const int mat  = rem / 12;               // 0 = We, 1 = W0
    const int tile = rem % 12;               // 16-column tile

    const float* __restrict__ src =
        (mat ? W0 : We) + (size_t)n * GRP_ELEMS + tile * 16;

    const int col = lane & 15;
    const int hi  = lane >> 4;               // 0: lanes 0-15, 1: lanes 16-31
    const int d0  = j * 256;

    // Incremented pointers: no per-iteration 64-bit index remat.
    const float* __restrict__ bsrc = src + (size_t)(d0 + hi * 2) * N1 + col;
    const float* __restrict__ xsrc = x + d0 + hi * 2;

    v8f   acc  = {};                         // C/D accumulator, starts at 0
    float sacc = 0.0f;                       // ||W0||^2 partial (mat==1)

    for (int s = 0; s < 64; ++s) {
        __builtin_prefetch((const void*)(bsrc + 16 * N1), 0, 3);

        v2f b;
        b.x = bsrc[0];                       // row d + hi*2
        b.y = bsrc[N1];                      // row d + hi*2 + 1

        // Unconditional broadcast loads, then lane-select (v_cndmask):
        // keeps EXEC all-ones, no divergent load blocks.
        const float ax = xsrc[0];
        const float ay = xsrc[1];
        v2f a;
        a.x = (col == 0) ? ax : 0.0f;
        a.y = (col == 0) ? ay : 0.0f;

        sacc += b.x * b.x + b.y * b.y;

        acc = __builtin_amdgcn_wmma_f32_16x16x4_f32(
            /*neg_a=*/false, a, /*neg_b=*/false, b,
            /*c_mod=*/(short)0, acc, /*reuse_a=*/false, /*reuse_b=*/false);

        bsrc += 4 * N1;
        xsrc += 4;
    }

    // D row 0, cols 0..15 live in acc[0] on lanes 0..15.
    float out = acc[0];

    // Tail: d = 1024 is the bias row, h1[1024] = 1.0 (handled by split j==3).
    if (j == 3 && lane < 16) {
        const float t = src[(size_t)D_IN * N1 + col];
        out  += t;
        sacc += t * t;
    }

    float* __restrict__ part = mat ? partP : partA;
    if (lane < 16)
        part[(size_t)(j * N2 + n) * N1 + tile * 16 + lane] = out;

    if (mat) {
        for (int m = 16; m >= 1; m >>= 1)
            sacc += __shfl_xor(sacc, m, 32);
        if (lane == 0)
            s0p[n * 48 + tile * 4 + j] = sacc;
    }
}

// ---------------------------------------------------------------------------
// Block-wide reduction over 192 threads (6 waves): wave shuffle + LDS.
// ---------------------------------------------------------------------------
__device__ __forceinline__ float block_reduce192(float v, float* lds)
{
    for (int m = 16; m >= 1; m >>= 1)
        v += __shfl_xor(v, m, 32);
    const int wave = threadIdx.x >> 5;
    if ((threadIdx.x & 31) == 0) lds[wave] = v;
    __syncthreads();
    const float s = lds[0] + lds[1] + lds[2] + lds[3] + lds[4] + lds[5];
    __syncthreads();
    return s;
}

// ---------------------------------------------------------------------------
// Stage 2: per group, reduce partials, normalize A, then run the exact
// collapsed 100-step recurrence on (c, v[192]). One block of 192 threads
// per group; thread t owns component k = t.
// ---------------------------------------------------------------------------
__global__ __launch_bounds__(192) void bls_recur(
    const float* __restrict__ x,
    const float* __restrict__ partA,
    const float* __restrict__ partP,
    const float* __restrict__ s0p,
    float* __restrict__ vOut,    // [48][192]
    float* __restrict__ cOut)    // [48]
{
    __shared__ float lds[8];
    const int n = blockIdx.x;
    const int t = threadIdx.x;

    // q = ||h1||^2 = sum(x^2) + 1
    float qs = 0.0f;
    for (int i = t; i < D_IN; i += 192) qs += x[i] * x[i];
    const float q = block_reduce192(qs, lds) + 1.0f;

    // s0 = ||W0_n||_F^2
    const float s0 = block_reduce192((t < 48) ? s0p[n * 48 + t] : 0.0f, lds);

    // A_raw, p  (sum the 4 d-split partials)
    float Araw = 0.0f, p = 0.0f;
    for (int jj = 0; jj < 4; ++jj) {
        Araw += partA[(size_t)(jj * N2 + n) * N1 + t];
        p    += partP[(size_t)(jj * N2 + n) * N1 + t];
    }
    const float na = sqrtf(block_reduce192(Araw * Araw, lds));
    const float A  = Araw / na;

    // Collapsed recurrence: W_t = c_t * W0 + h1 * v_t^T
    float c = 1.0f, v = 0.0f;
    for (int it = 0; it < ITRS; ++it) {
        const float pv = block_reduce192(p * v, lds);
        const float v2 = block_reduce192(v * v, lds);
        const float nt = sqrtf(c * c * s0 + 2.0f * c * pv + q * v2);
        const float f  = 1.0f - (LR * LAM) / nt;
        const float r  = c * p + q * v - A;
        v = f * v - 2.0f * LR * r;
        c = f * c;
    }

    vOut[(size_t)n * N1 + t] = v;
    if (t == 0) cOut[n] = c;
}

// ---------------------------------------------------------------------------
// Stage 3: W_all[n,d,k] = c[n]*W0[n,d,k] + h1[d]*v[n,k], float4-vectorized
// (k fastest, N1 % 4 == 0 so a float4 never crosses a d boundary).
// ---------------------------------------------------------------------------
__global__ __launch_bounds__(256) void bls_writeout(
    const float* __restrict__ x,
    const float* __restrict__ W0,
    const float* __restrict__ vIn,
    const float* __restrict__ cIn,
    float* __restrict__ out)
{
    const size_t i4   = (size_t)blockIdx.x * 256 + threadIdx.x;
    const size_t base = i4 * 4;
    const int n   = (int)(base / GRP_ELEMS);
    const int rem = (int)(base - (size_t)n * GRP_ELEMS);
    const int d   = rem / N1;
    const int k   = rem % N1;

    const float4 w  = *(const float4*)(W0 + base);
    const float4 vv = *(const float4*)(vIn + (size_t)n * N1 + k);
    const float  h  = (d < D_IN) ? x[d] : 1.0f;
    const float  cc = cIn[n];

    float4 o;
    o.x = cc * w.x + h * vv.x;
    o.y = cc * w.y + h * vv.y;
    o.z = cc * w.z + h * vv.z;
    o.w = cc * w.w + h * vv.w;
    *(float4*)(out + base) = o;
}

// ---------------------------------------------------------------------------
// Stage 4: pred = x @ Wp^T + bp.  One wave per output row (512 rows).
// ---------------------------------------------------------------------------
__global__ __launch_bounds__(256) void bls_pred(
    const float* __restrict__ x,
    const float* __restrict__ Wp,
    const float* __restrict__ bp,
    float* __restrict__ out)
{
    const int wave = (blockIdx.x * 256 + threadIdx.x) >> 5;   // 0..511
    const int lane = threadIdx.x & 31;
    const float* __restrict__ row = Wp + (size_t)wave * D_IN;

    float acc = 0.0f;
    for (int i = lane; i < D_IN; i += 32)
        acc += x[i] * row[i];
    for (int m = 16; m >= 1; m >>= 1)
        acc += __shfl_xor(acc, m, 32);
    if (lane == 0)
        out[wave] = acc + bp[wave];
}

// ---------------------------------------------------------------------------
// Launch. d_in: x, We, W0, Wp, bp. d_out: [pred(512) | W_all(9,446,400)].
// ws usage: 4*48*192 *2 partials + 48*48 s0 partials + 48*192 v + 48 c
//         = 85,296 floats (~341 KB).
// ---------------------------------------------------------------------------
extern "C" void kernel_launch(void* const* d_in, const int* in_sizes, int n_in,
                              void* d_out, int out_size, void* d_ws, size_t ws_size,
                              hipStream_t stream)
{
    const float* x  = (const float*)d_in[0];
    const float* We = (const float*)d_in[1];
    const float* W0 = (const float*)d_in[2];
    const float* Wp = (const float*)d_in[3];
    const float* bp = (const float*)d_in[4];
    float* out = (float*)d_out;

    float* ws    = (float*)d_ws;
    float* partA = ws;                         // 4*48*192 = 36864
    float* partP = partA + 4 * N2 * N1;        // 36864
    float* s0p   = partP + 4 * N2 * N1;        // 2304
    float* vOut  = s0p + N2 * 48;              // 9216
    float* cOut  = vOut + N2 * N1;             // 48

    dim3 g1(N2 * 2 * 12, 4);                   // 1152 x 4 single-wave blocks
    bls_matvec_wmma<<<g1, 32, 0, stream>>>(x, We, W0, partA, partP, s0p);

    bls_recur<<<N2, N1, 0, stream>>>(x, partA, partP, s0p, vOut, cOut);

    bls_writeout<<<TOT_W / (256 * 4), 256, 0, stream>>>(x, W0, vOut, cOut,
                                                        out + D_OUT);

    bls_pred<<<D_OUT / 8, 256, 0, stream>>>(x, Wp, bp, out);   // 64 blocks = 512 waves
}